// Attn_88725434401526
// MI455X (gfx1250) — compile-verified
//
#include <hip/hip_runtime.h>

typedef __attribute__((ext_vector_type(2))) float v2f;
typedef __attribute__((ext_vector_type(4))) float v4f;
typedef __attribute__((ext_vector_type(8))) float v8f;

#define BATCH 16
#define LOCM  20000
#define MLEN  100
#define EMB   256
#define WAVES_PER_WG 10
#define LDS_STRIDE 260   // 256 + 4 floats pad: banks (4n + c) % 64 distinct for n=0..15

// out[b,l] = sum_m ( Ecand[l,:] . attn[b,m,:] ) * (sd[b,m,l,0]+sd[b,m,l,1]) * w[m]
__global__ __launch_bounds__(32 * WAVES_PER_WG)
void attn_delta_wmma(const float* __restrict__ attn,    // (16,100,256)
                     const float* __restrict__ sdelta,  // (16,100,20000,2)
                     const float* __restrict__ emb,     // (20001,256)
                     const float* __restrict__ vw,      // (100,)
                     float* __restrict__ out)           // (16,20000)
{
    __shared__ float panel[16 * LDS_STRIDE];

    const int tid  = threadIdx.x;
    const int lane = tid & 31;
    const int wave = tid >> 5;
    const int n    = lane & 15;   // A row-in-tile / B column-in-tile
    const int kh   = lane >> 4;   // K-half (ISA fp32 16x16x4 operand layout)

    const int ltile = blockIdx.x * WAVES_PER_WG + wave;  // 0..1249, exact (1250*16 = 20000)
    const int l0    = ltile * 16;
    const int b0    = blockIdx.y * 8;                    // grid.y == 2

    // ---- preload A panel: Ecand[l0+n, :] ; per k-step ks lane holds cols 4*ks+2*kh, +1
    const float* Erow = emb + (size_t)(1 + l0 + n) * EMB;   // skip emb_table row 0
    v2f apan[64];
#pragma unroll
    for (int ks = 0; ks < 64; ++ks)
        apan[ks] = *(const v2f*)(Erow + 4 * ks + 2 * kh);

    for (int bb = 0; bb < 8; ++bb) {
        const int b = b0 + bb;
        v8f acc = {};
        for (int mt = 0; mt < 7; ++mt) {
            const int m0 = mt * 16;
            __syncthreads();
            // stage B panel: attn[b, m0..m0+15 (clamped), 0..255] into LDS (padded rows)
            for (int idx = tid; idx < 16 * 64; idx += 32 * WAVES_PER_WG) {
                const int row = idx >> 6;
                const int c4  = idx & 63;
                const int mr0 = m0 + row;
                const int mr  = (mr0 < MLEN) ? mr0 : (MLEN - 1);
                v4f v = *(const v4f*)(attn + ((size_t)b * MLEN + mr) * EMB + c4 * 4);
                *(v4f*)&panel[row * LDS_STRIDE + c4 * 4] = v;
            }
            __syncthreads();

            // P(16l x 16m) = A(16x256) x B(256x16): 64 chained fp32 WMMAs over K
            v8f c = {};
#pragma unroll
            for (int ks = 0; ks < 64; ++ks) {
                const v2f bv = *(const v2f*)&panel[n * LDS_STRIDE + 4 * ks + 2 * kh];
                c = __builtin_amdgcn_wmma_f32_16x16x4_f32(
                        false, apan[ks], false, bv, (short)0, c, false, false);
            }

            // elementwise delta2 * w[m], accumulate over m-tiles
            const int  m     = m0 + n;
            const bool valid = (m < MLEN);
            const float wm   = valid ? vw[m] : 0.0f;
            const int  mc    = valid ? m : (MLEN - 1);
            // 64 contiguous bytes per lane: pairs (d0,d1) for l = l0+8*kh .. +7
            const float* dptr = sdelta + ((size_t)(b * MLEN + mc) * LOCM + (l0 + 8 * kh)) * 2;
            v4f q0 = __builtin_nontemporal_load((const v4f*)(dptr + 0));
            v4f q1 = __builtin_nontemporal_load((const v4f*)(dptr + 4));
            v4f q2 = __builtin_nontemporal_load((const v4f*)(dptr + 8));
            v4f q3 = __builtin_nontemporal_load((const v4f*)(dptr + 12));
            const float d2[8] = { q0.x + q0.y, q0.z + q0.w,
                                  q1.x + q1.y, q1.z + q1.w,
                                  q2.x + q2.y, q2.z + q2.w,
                                  q3.x + q3.y, q3.z + q3.w };
#pragma unroll
            for (int v = 0; v < 8; ++v)
                acc[v] = fmaf(c[v], d2[v] * wm, acc[v]);
        }

        // reduce the 16 m-columns: butterfly within each 16-lane half
#pragma unroll
        for (int v = 0; v < 8; ++v) {
            float a = acc[v];
            a += __shfl_xor(a, 1, 16);
            a += __shfl_xor(a, 2, 16);
            a += __shfl_xor(a, 4, 16);
            a += __shfl_xor(a, 8, 16);
            acc[v] = a;
        }
        if (n == 0) {  // lane 0 -> l0..l0+7, lane 16 -> l0+8..l0+15
            float* op = out + (size_t)b * LOCM + l0 + 8 * kh;
#pragma unroll
            for (int v = 0; v < 8; ++v) op[v] = acc[v];
        }
    }
}

extern "C" void kernel_launch(void* const* d_in, const int* in_sizes, int n_in,
                              void* d_out, int out_size, void* d_ws, size_t ws_size,
                              hipStream_t stream) {
    const float* attn   = (const float*)d_in[0];
    const float* sdelta = (const float*)d_in[1];
    // d_in[2] = traj_len: unused by the reference output
    const float* emb    = (const float*)d_in[3];
    const float* vw     = (const float*)d_in[4];
    float* out = (float*)d_out;

    dim3 grid(125, 2, 1);          // 125 WGs x 10 waves = 1250 l-tiles; y splits batch
    dim3 block(32 * WAVES_PER_WG); // 320 threads = 10 wave32
    attn_delta_wmma<<<grid, block, 0, stream>>>(attn, sdelta, emb, vw, out);
}